// TopKRouter_19464791786098
// MI455X (gfx1250) — compile-verified
//
#include <hip/hip_runtime.h>
#include <math.h>

typedef __attribute__((ext_vector_type(2))) float v2f;
typedef __attribute__((ext_vector_type(8))) float v8f;

#define KC    32     // K-chunk staged in LDS
#define PAD   4      // LDS row pad: stride 36 -> conflict-free fragment reads
#define TPB   256
#define MTILE 64     // tokens per block
#define ETILE 64     // experts (E)

__global__ __launch_bounds__(TPB)
void router_gemm_topk(const float* __restrict__ x,
                      const float* __restrict__ W,
                      const float* __restrict__ bias,
                      float* __restrict__ outp,
                      int* __restrict__ outidx,
                      int D)
{
    __shared__ float xs[2][MTILE][KC + PAD];
    __shared__ float ws[2][ETILE][KC + PAD];
    __shared__ float lg[MTILE][ETILE + 1];   // stride 65 -> conflict-free scan
    __shared__ float tv[MTILE][2];
    __shared__ int   ti[MTILE][2];

    const int tid  = threadIdx.x;
    const int lane = tid & 31;
    const int wave = tid >> 5;
    const int wm   = wave >> 1;     // 0..3 : 16-token sub-tile
    const int wn   = wave & 1;      // 0..1 : 32-expert sub-tile (2 accumulators)
    const int m0   = blockIdx.x * MTILE;

    // global->LDS mapping: 64 rows, 4 threads/row, 2 float4 per thread per array
    const int grow = tid >> 2;
    const int gq   = tid & 3;

    const float* xgr = x + (size_t)(m0 + grow) * (size_t)D;
    const float* wgr = W + (size_t)grow * (size_t)D;

    float4 xr[2], wr[2];
    const int nChunks = D / KC;

    // ---- prefetch + stage chunk 0 ----
    {
        const float4* px = (const float4*)xgr;
        const float4* pw = (const float4*)wgr;
#pragma unroll
        for (int i = 0; i < 2; ++i) { xr[i] = px[gq + i * 4]; wr[i] = pw[gq + i * 4]; }
    }
#pragma unroll
    for (int i = 0; i < 2; ++i) {
        int q = gq + i * 4;
        xs[0][grow][q * 4 + 0] = xr[i].x; xs[0][grow][q * 4 + 1] = xr[i].y;
        xs[0][grow][q * 4 + 2] = xr[i].z; xs[0][grow][q * 4 + 3] = xr[i].w;
        ws[0][grow][q * 4 + 0] = wr[i].x; ws[0][grow][q * 4 + 1] = wr[i].y;
        ws[0][grow][q * 4 + 2] = wr[i].z; ws[0][grow][q * 4 + 3] = wr[i].w;
    }
    __syncthreads();

    // WMMA f32 16x16x4 fragment addressing:
    // A 16x4: lanes 0-15 rows with K {k,k+1}; lanes 16-31 same rows, K {k+2,k+3}
    // B 4x16: lanes 0-15 cols with K {k,k+1}; lanes 16-31 cols with K {k+2,k+3}
    v8f c0 = {}, c1 = {};
    const int arow  = wm * 16 + (lane & 15);
    const int brow0 = wn * 32 + (lane & 15);
    const int brow1 = brow0 + 16;
    const int koff  = (lane >> 4) << 1;

    for (int kc = 0; kc < nChunks; ++kc) {
        const int cur = kc & 1;
        if (kc + 1 < nChunks) {   // register prefetch of next chunk
            const float4* px = (const float4*)(xgr + (size_t)(kc + 1) * KC);
            const float4* pw = (const float4*)(wgr + (size_t)(kc + 1) * KC);
#pragma unroll
            for (int i = 0; i < 2; ++i) { xr[i] = px[gq + i * 4]; wr[i] = pw[gq + i * 4]; }
        }

        const float* xbase  = &xs[cur][arow][koff];
        const float* wbase0 = &ws[cur][brow0][koff];
        const float* wbase1 = &ws[cur][brow1][koff];
#pragma unroll
        for (int kk = 0; kk < KC; kk += 4) {
            v2f a  = *(const v2f*)(xbase  + kk);   // shared A fragment
            v2f b0 = *(const v2f*)(wbase0 + kk);
            v2f b1 = *(const v2f*)(wbase1 + kk);
            c0 = __builtin_amdgcn_wmma_f32_16x16x4_f32(false, a, false, b0,
                                                       (short)0, c0, false, false);
            c1 = __builtin_amdgcn_wmma_f32_16x16x4_f32(false, a, false, b1,
                                                       (short)0, c1, false, false);
        }
        __syncthreads();

        if (kc + 1 < nChunks) {
            const int nxt = (kc + 1) & 1;
#pragma unroll
            for (int i = 0; i < 2; ++i) {
                int q = gq + i * 4;
                xs[nxt][grow][q * 4 + 0] = xr[i].x; xs[nxt][grow][q * 4 + 1] = xr[i].y;
                xs[nxt][grow][q * 4 + 2] = xr[i].z; xs[nxt][grow][q * 4 + 3] = xr[i].w;
                ws[nxt][grow][q * 4 + 0] = wr[i].x; ws[nxt][grow][q * 4 + 1] = wr[i].y;
                ws[nxt][grow][q * 4 + 2] = wr[i].z; ws[nxt][grow][q * 4 + 3] = wr[i].w;
            }
            __syncthreads();
        }
    }

    // ---- epilogue: bias add, stash logits ----
    // C/D layout: VGPR i -> M = i (lanes 0-15) or 8+i (lanes 16-31); N = lane&15
    const int ncol0 = wn * 32 + (lane & 15);
    const int ncol1 = ncol0 + 16;
    const float bv0 = bias[ncol0];
    const float bv1 = bias[ncol1];
    const int mbase = wm * 16 + ((lane >> 4) << 3);
#pragma unroll
    for (int i = 0; i < 8; ++i) {
        lg[mbase + i][ncol0] = c0[i] + bv0;
        lg[mbase + i][ncol1] = c1[i] + bv1;
    }
    __syncthreads();

    // ---- top-2 + exact 2-way softmax (one lane per token) ----
    if (tid < MTILE) {
        float v1 = -INFINITY, v2 = -INFINITY;
        int   i1 = 0, i2 = 0;
        for (int e = 0; e < ETILE; ++e) {
            float v = lg[tid][e];
            if (v > v1)      { v2 = v1; i2 = i1; v1 = v; i1 = e; }
            else if (v > v2) { v2 = v;  i2 = e; }
        }
        float e2  = __expf(v2 - v1);
        float inv = 1.0f / (1.0f + e2);
        tv[tid][0] = inv;        // p(top1)
        tv[tid][1] = e2 * inv;   // p(top2)
        ti[tid][0] = i1;
        ti[tid][1] = i2;
        size_t g = (size_t)(m0 + tid);
        outidx[g * 2 + 0] = i1;
        outidx[g * 2 + 1] = i2;
    }
    __syncthreads();

    // ---- coalesced sparse-softmax scatter ----
#pragma unroll
    for (int j = 0; j < (MTILE * ETILE) / TPB; ++j) {
        int linear = tid + j * TPB;
        int t = linear >> 6;
        int e = linear & 63;
        int i1 = ti[t][0], i2 = ti[t][1];
        float v = (e == i1) ? tv[t][0] : ((e == i2) ? tv[t][1] : 0.0f);
        outp[(size_t)(m0 + t) * ETILE + e] = v;
    }
}

extern "C" void kernel_launch(void* const* d_in, const int* in_sizes, int n_in,
                              void* d_out, int out_size, void* d_ws, size_t ws_size,
                              hipStream_t stream) {
    const float* x = (const float*)d_in[0];
    const float* W = (const float*)d_in[1];
    const float* b = (const float*)d_in[2];

    const int E      = in_sizes[2];               // 64
    const int D      = in_sizes[1] / E;           // 2048
    const int tokens = in_sizes[0] / D;           // 16384

    float* outp   = (float*)d_out;
    int*   outidx = (int*)((float*)d_out + (size_t)tokens * (size_t)E);

    const int blocks = tokens / MTILE;            // 256
    hipLaunchKernelGGL(router_gemm_topk, dim3(blocks), dim3(TPB), 0, stream,
                       x, W, b, outp, outidx, D);
}